// AttentionEncoder_56916906607265
// MI455X (gfx1250) — compile-verified
//
#include <hip/hip_runtime.h>
#include <hip/hip_bf16.h>
#include <math.h>

// ---------------- problem constants ----------------
constexpr int Bb = 64, Tt = 64, Nn = 512, Fd = 16, Mm = 128, Ee = 8192;
constexpr int EN = Ee + Nn;   // edges + self loops = 8704
constexpr int G4 = 4 * Mm;    // 512

// ---------------- workspace layout (floats) ----------------
constexpr size_t OFF_XGAT = 0;                                   // (B,N,T)
constexpr size_t OFF_H    = OFF_XGAT + (size_t)Bb * Nn * Tt;     // (B,N,T)
constexpr size_t OFF_GOUT = OFF_H    + (size_t)Bb * Nn * Tt;     // (B,N,T)
constexpr size_t OFF_Y    = OFF_GOUT + (size_t)Bb * Nn * Tt;     // (B,N,T)
constexpr size_t OFF_AS   = OFF_Y    + (size_t)Bb * Nn * Tt;     // (B,N)
constexpr size_t OFF_AD   = OFF_AS   + (size_t)Bb * Nn;
constexpr size_t OFF_MAX  = OFF_AD   + (size_t)Bb * Nn;
constexpr size_t OFF_DEN  = OFF_MAX  + (size_t)Bb * Nn;
constexpr size_t OFF_Q    = OFF_DEN  + (size_t)Bb * Nn;          // (B,T)
constexpr size_t OFF_EK   = OFF_Q    + (size_t)Bb * Tt;          // (B,N)
constexpr size_t OFF_WIN  = OFF_EK   + (size_t)Bb * Nn;          // (B,N)
constexpr size_t OFF_HST  = OFF_WIN  + (size_t)Bb * Nn;          // (B,M)
constexpr size_t OFF_CST  = OFF_HST  + (size_t)Bb * Mm;          // (B,M)
constexpr size_t OFF_GATE = OFF_CST  + (size_t)Bb * Mm;          // (B,4M)

// ---------------- CDNA5 f32 WMMA 16x16x4 ----------------
typedef __attribute__((ext_vector_type(2))) float v2f;
typedef __attribute__((ext_vector_type(8))) float v8f;

#if defined(__has_builtin)
#if __has_builtin(__builtin_amdgcn_wmma_f32_16x16x4_f32)
#define HAVE_WMMA_F32 1
#endif
#endif

__device__ __forceinline__ v8f wmma4(v2f a, v2f b, v8f c) {
#ifdef HAVE_WMMA_F32
  // (neg_a, A, neg_b, B, c_mod, C, reuse_a, reuse_b)
  return __builtin_amdgcn_wmma_f32_16x16x4_f32(false, a, false, b, (short)0, c,
                                               false, false);
#else
  // compile-safety fallback only (histogram will show wmma==0)
  #pragma unroll
  for (int i = 0; i < 8; ++i) c[i] = fmaf(a.x, b.x, fmaf(a.y, b.y, c[i]));
  return c;
#endif
}

__device__ __forceinline__ float sigf(float x) { return 1.f / (1.f + expf(-x)); }

__device__ __forceinline__ void atomicMaxF(float* addr, float val) {
  unsigned* ua = (unsigned*)addr;
  unsigned old = *ua;
  while (__uint_as_float(old) < val) {
    unsigned assumed = old;
    old = atomicCAS(ua, assumed, __float_as_uint(val));
    if (old == assumed) break;
  }
}

// ---------------- stage 1: new_input -> x_gat (B,N,T) ----------------
__global__ void k_proj(const float* __restrict__ inp, const float* __restrict__ A_w,
                       const float* __restrict__ A_b, float* __restrict__ x_gat) {
  int gid = blockIdx.x * blockDim.x + threadIdx.x;    // (b*N+n)*T + t
  int t = gid % Tt;
  int n = (gid / Tt) % Nn;
  int b = gid / (Nn * Tt);
  const float* p = inp + (size_t)((b * Tt + t) * Nn + n) * Fd;
  float s = A_b[0];
  #pragma unroll
  for (int f = 0; f < Fd; ++f) s = fmaf(p[f], A_w[f], s);
  x_gat[gid] = s;
}

// init gat_out with bias, segmax/-inf, den/0, h0/c0 = 0
__global__ void k_init(const float* __restrict__ gat_b, float* __restrict__ gout,
                       float* __restrict__ segmax, float* __restrict__ den,
                       float* __restrict__ h_st, float* __restrict__ c_st) {
  int gid = blockIdx.x * blockDim.x + threadIdx.x;    // B*N*T
  gout[gid] = gat_b[gid & 63];
  if (gid < Bb * Nn) { segmax[gid] = -3.0e38f; den[gid] = 0.f; }
  if (gid < Bb * Mm) { h_st[gid] = 0.f; c_st[gid] = 0.f; }
}

// ---------------- GEMM: D(R,64) = A(R,64) @ W(64,64)^T via f32 WMMA ----------------
// one wave per 16x16 tile, 4 waves/block; grid = Rtiles blocks (exact -> EXEC all 1s)
__global__ void k_gemm64(const float* __restrict__ Amat, const float* __restrict__ Wmat,
                         float* __restrict__ D) {
  int lane = threadIdx.x & 31;
  int wave = blockIdx.x * 4 + (threadIdx.x >> 5);
  int tileR = wave >> 2, tileC = wave & 3;
  int m = lane & 15, koff = (lane >> 4) << 1;
  const float* arow = Amat + (size_t)(tileR * 16 + m) * 64;
  const float* wrow = Wmat + (size_t)(tileC * 16 + m) * 64;
  v8f acc = {};
  #pragma unroll
  for (int k0 = 0; k0 < 64; k0 += 4) {
    v2f a, w;
    a.x = arow[k0 + koff]; a.y = arow[k0 + koff + 1];
    w.x = wrow[k0 + koff]; w.y = wrow[k0 + koff + 1];
    acc = wmma4(a, w, acc);
  }
  int col = tileC * 16 + m;
  int rbase = tileR * 16 + ((lane >> 4) << 3);
  #pragma unroll
  for (int r = 0; r < 8; ++r) D[(size_t)(rbase + r) * 64 + col] = acc[r];
}

// ---------------- dual-K GEMM: D = A1@W1^T + A2@W2^T + bias ----------------
__global__ void k_gemm_dual(const float* __restrict__ A1, int lda1, int K1,
                            const float* __restrict__ W1, int ldw1,
                            const float* __restrict__ A2, int lda2, int K2,
                            const float* __restrict__ W2, int ldw2,
                            const float* __restrict__ bias1,
                            const float* __restrict__ bias2,
                            float* __restrict__ D, int ldd, int Ctiles) {
  int lane = threadIdx.x & 31;
  int wave = blockIdx.x * 4 + (threadIdx.x >> 5);
  int tileR = wave / Ctiles, tileC = wave % Ctiles;
  int m = lane & 15, koff = (lane >> 4) << 1;
  v8f acc = {};
  {
    const float* ar = A1 + (size_t)(tileR * 16 + m) * lda1;
    const float* wr = W1 + (size_t)(tileC * 16 + m) * ldw1;
    for (int k0 = 0; k0 < K1; k0 += 4) {
      v2f a, w;
      a.x = ar[k0 + koff]; a.y = ar[k0 + koff + 1];
      w.x = wr[k0 + koff]; w.y = wr[k0 + koff + 1];
      acc = wmma4(a, w, acc);
    }
  }
  {
    const float* ar = A2 + (size_t)(tileR * 16 + m) * lda2;
    const float* wr = W2 + (size_t)(tileC * 16 + m) * ldw2;
    for (int k0 = 0; k0 < K2; k0 += 4) {
      v2f a, w;
      a.x = ar[k0 + koff]; a.y = ar[k0 + koff + 1];
      w.x = wr[k0 + koff]; w.y = wr[k0 + koff + 1];
      acc = wmma4(a, w, acc);
    }
  }
  int col = tileC * 16 + m;
  float bv = bias1[col] + (bias2 ? bias2[col] : 0.f);
  int rbase = tileR * 16 + ((lane >> 4) << 3);
  #pragma unroll
  for (int r = 0; r < 8; ++r) D[(size_t)(rbase + r) * ldd + col] = acc[r] + bv;
}

// ---------------- GAT attention scalars ----------------
__global__ void k_att_sd(const float* __restrict__ hbuf, const float* __restrict__ att_s,
                         const float* __restrict__ att_d, float* __restrict__ a_s,
                         float* __restrict__ a_d) {
  int gid = blockIdx.x * blockDim.x + threadIdx.x;    // B*N
  const float* hp = hbuf + (size_t)gid * 64;
  float ss = 0.f, sd = 0.f;
  #pragma unroll
  for (int t = 0; t < 64; ++t) {
    float hv = hp[t];
    ss = fmaf(hv, att_s[t], ss);
    sd = fmaf(hv, att_d[t], sd);
  }
  a_s[gid] = ss; a_d[gid] = sd;
}

__device__ __forceinline__ void edge_decode(const int* __restrict__ ei, int e,
                                            int& src, int& dst) {
  if (e < Ee) { src = ei[e]; dst = ei[Ee + e]; }
  else        { src = e - Ee; dst = e - Ee; }
}

__global__ void k_edge_max(const int* __restrict__ ei, const float* __restrict__ a_s,
                           const float* __restrict__ a_d, float* __restrict__ segmax) {
  int gid = blockIdx.x * blockDim.x + threadIdx.x;
  if (gid >= Bb * EN) return;
  int b = gid / EN, e = gid % EN, src, dst;
  edge_decode(ei, e, src, dst);
  float x = a_s[b * Nn + src] + a_d[b * Nn + dst];
  float ev = x > 0.f ? x : 0.2f * x;                  // leaky_relu(0.2)
  atomicMaxF(&segmax[b * Nn + dst], ev);
}

__global__ void k_edge_sum(const int* __restrict__ ei, const float* __restrict__ a_s,
                           const float* __restrict__ a_d, const float* __restrict__ segmax,
                           float* __restrict__ den) {
  int gid = blockIdx.x * blockDim.x + threadIdx.x;
  if (gid >= Bb * EN) return;
  int b = gid / EN, e = gid % EN, src, dst;
  edge_decode(ei, e, src, dst);
  float x = a_s[b * Nn + src] + a_d[b * Nn + dst];
  float ev = x > 0.f ? x : 0.2f * x;
  atomicAdd(&den[b * Nn + dst], expf(ev - segmax[b * Nn + dst]));
}

__global__ void k_edge_scatter(const int* __restrict__ ei, const float* __restrict__ a_s,
                               const float* __restrict__ a_d,
                               const float* __restrict__ segmax,
                               const float* __restrict__ den,
                               const float* __restrict__ hbuf, float* __restrict__ gout) {
  int gid = blockIdx.x * blockDim.x + threadIdx.x;
  if (gid >= Bb * EN) return;
  int b = gid / EN, e = gid % EN, src, dst;
  edge_decode(ei, e, src, dst);
  float x = a_s[b * Nn + src] + a_d[b * Nn + dst];
  float ev = x > 0.f ? x : 0.2f * x;
  float alpha = expf(ev - segmax[b * Nn + dst]) / den[b * Nn + dst];
  const float4* h4 = (const float4*)(hbuf + (size_t)(b * Nn + src) * 64);
  float* go = gout + (size_t)(b * Nn + dst) * 64;
  #pragma unroll
  for (int t4 = 0; t4 < 16; ++t4) {
    float4 hv = h4[t4];
    atomicAdd(go + t4 * 4 + 0, alpha * hv.x);
    atomicAdd(go + t4 * 4 + 1, alpha * hv.y);
    atomicAdd(go + t4 * 4 + 2, alpha * hv.z);
    atomicAdd(go + t4 * 4 + 3, alpha * hv.w);
  }
}

// ---------------- scan step: attention energies ek[b,n] ----------------
__global__ void k_ek(const float* __restrict__ q, const float* __restrict__ ybuf,
                     const float* __restrict__ v_e, float* __restrict__ ek) {
  __shared__ float qs[64], vs[64];
  int b = blockIdx.y;
  int n = blockIdx.x * 256 + threadIdx.x;
  if (threadIdx.x < 64) {
    qs[threadIdx.x] = q[b * 64 + threadIdx.x];
    vs[threadIdx.x] = v_e[threadIdx.x];
  }
  __syncthreads();
  const float* yp = ybuf + (size_t)(b * Nn + n) * 64;
  float acc = 0.f;
  #pragma unroll
  for (int t = 0; t < 64; ++t) acc = fmaf(tanhf(qs[t] + yp[t]), vs[t], acc);
  ek[b * Nn + n] = acc;
}

// softmax over N, then w_in[b,n] = alpha * gat_out[b,n,t]   (x_t == new2[:,t,:])
__global__ void k_softmax_win(const float* __restrict__ ek, const float* __restrict__ gout,
                              float* __restrict__ w_in, int t) {
  __shared__ float red[256];
  int b = blockIdx.x, tid = threadIdx.x;
  float v0 = ek[b * Nn + tid], v1 = ek[b * Nn + 256 + tid];
  red[tid] = fmaxf(v0, v1);
  __syncthreads();
  for (int s = 128; s > 0; s >>= 1) {
    if (tid < s) red[tid] = fmaxf(red[tid], red[tid + s]);
    __syncthreads();
  }
  float mx = red[0];
  __syncthreads();
  float e0 = expf(v0 - mx), e1 = expf(v1 - mx);
  red[tid] = e0 + e1;
  __syncthreads();
  for (int s = 128; s > 0; s >>= 1) {
    if (tid < s) red[tid] += red[tid + s];
    __syncthreads();
  }
  float inv = 1.f / red[0];
  w_in[b * Nn + tid]       = e0 * inv * gout[(size_t)(b * Nn + tid) * 64 + t];
  w_in[b * Nn + 256 + tid] = e1 * inv * gout[(size_t)(b * Nn + 256 + tid) * 64 + t];
}

// LSTM pointwise update; writes hs[b,t,:]
__global__ void k_lstm(const float* __restrict__ gates, float* __restrict__ h_st,
                       float* __restrict__ c_st, float* __restrict__ out, int t) {
  int gid = blockIdx.x * blockDim.x + threadIdx.x;   // B*M
  int b = gid >> 7, m = gid & 127;
  const float* g = gates + (size_t)b * G4;
  float gi = g[m], gf = g[128 + m], gg = g[256 + m], go = g[384 + m];
  float c = c_st[gid];
  float cn = sigf(gf) * c + sigf(gi) * tanhf(gg);
  float hn = sigf(go) * tanhf(cn);
  c_st[gid] = cn;
  h_st[gid] = hn;
  out[((size_t)b * Tt + t) * Mm + m] = hn;
}

// ---------------- launch ----------------
extern "C" void kernel_launch(void* const* d_in, const int* in_sizes, int n_in,
                              void* d_out, int out_size, void* d_ws, size_t ws_size,
                              hipStream_t stream) {
  const float* inputs = (const float*)d_in[0];
  const int*   eidx   = (const int*)d_in[1];
  const float* A_w    = (const float*)d_in[2];
  const float* A_b    = (const float*)d_in[3];
  const float* gat_w  = (const float*)d_in[4];
  const float* att_s  = (const float*)d_in[5];
  const float* att_d  = (const float*)d_in[6];
  const float* gat_b  = (const float*)d_in[7];
  const float* W_e_w  = (const float*)d_in[8];
  const float* W_e_b  = (const float*)d_in[9];
  const float* U_e_w  = (const float*)d_in[10];
  const float* v_e_w  = (const float*)d_in[11];
  const float* W_ih   = (const float*)d_in[12];
  const float* W_hh   = (const float*)d_in[13];
  const float* b_ih   = (const float*)d_in[14];
  const float* b_hh   = (const float*)d_in[15];
  (void)in_sizes; (void)n_in; (void)out_size; (void)ws_size;

  float* out   = (float*)d_out;
  float* ws    = (float*)d_ws;
  float* x_gat = ws + OFF_XGAT;
  float* hbuf  = ws + OFF_H;
  float* gout  = ws + OFF_GOUT;
  float* ybuf  = ws + OFF_Y;
  float* a_s   = ws + OFF_AS;
  float* a_d   = ws + OFF_AD;
  float* smax  = ws + OFF_MAX;
  float* den   = ws + OFF_DEN;
  float* q     = ws + OFF_Q;
  float* ek    = ws + OFF_EK;
  float* w_in  = ws + OFF_WIN;
  float* h_st  = ws + OFF_HST;
  float* c_st  = ws + OFF_CST;
  float* gates = ws + OFF_GATE;

  // 1) new_input (transposed into (B,N,T))
  k_proj<<<Bb * Nn * Tt / 256, 256, 0, stream>>>(inputs, A_w, A_b, x_gat);
  // 2) GAT linear: h = x @ gat_w^T   (batched as one (B*N,64)x(64,64) GEMM)
  k_gemm64<<<Bb * Nn / 16, 128, 0, stream>>>(x_gat, gat_w, hbuf);
  // 3) attention scalars
  k_att_sd<<<Bb * Nn / 256, 256, 0, stream>>>(hbuf, att_s, att_d, a_s, a_d);
  // 4) init gat_out (+bias), seg buffers, LSTM state
  k_init<<<Bb * Nn * Tt / 256, 256, 0, stream>>>(gat_b, gout, smax, den, h_st, c_st);
  // 5) edge softmax + scatter
  int egrid = (Bb * EN + 255) / 256;
  k_edge_max<<<egrid, 256, 0, stream>>>(eidx, a_s, a_d, smax);
  k_edge_sum<<<egrid, 256, 0, stream>>>(eidx, a_s, a_d, smax, den);
  k_edge_scatter<<<egrid, 256, 0, stream>>>(eidx, a_s, a_d, smax, den, hbuf, gout);
  // 6) hoisted y = gat_out @ U_e_w^T
  k_gemm64<<<Bb * Nn / 16, 128, 0, stream>>>(gout, U_e_w, ybuf);

  // 7) sequential LSTM scan over T
  for (int t = 0; t < Tt; ++t) {
    // q = [h|c] @ W_e_w^T + W_e_b   (4x4 tiles = 16 waves)
    k_gemm_dual<<<4, 128, 0, stream>>>(h_st, Mm, Mm, W_e_w, 2 * Mm,
                                       c_st, Mm, Mm, W_e_w + Mm, 2 * Mm,
                                       W_e_b, nullptr, q, Tt, 4);
    k_ek<<<dim3(Nn / 256, Bb), 256, 0, stream>>>(q, ybuf, v_e_w, ek);
    k_softmax_win<<<Bb, 256, 0, stream>>>(ek, gout, w_in, t);
    // gates = w_in @ W_ih^T + h @ W_hh^T + b_ih + b_hh   (4x32 tiles = 128 waves)
    k_gemm_dual<<<32, 128, 0, stream>>>(w_in, Nn, Nn, W_ih, Nn,
                                        h_st, Mm, Mm, W_hh, Mm,
                                        b_ih, b_hh, gates, G4, 32);
    k_lstm<<<Bb * Mm / 256, 256, 0, stream>>>(gates, h_st, c_st, out, t);
  }
}